// TranscriptEmission_2052994367850
// MI455X (gfx1250) — compile-verified
//
#include <hip/hip_runtime.h>
#include <cstdint>

#define NSTATES 12
#define NK      200
#define TABLE_N (NSTATES * NK)   // 2400 floats = 9600 bytes

// ---------------------------------------------------------------------------
// NB log-pmf for (state s, integer count k) — matches reference order in f32.
// ---------------------------------------------------------------------------
__device__ __forceinline__ float nb_logp(float log_mean, float log_frac_std, int k) {
    float means = expf(log_mean);
    float frac  = expf(log_frac_std);
    float mf    = means * frac;
    float var   = means + mf * mf;
    float ps    = means / var;
    float rs    = (means * means) / (var - means);
    float ts    = rs / ps;
    float fs    = ts - rs;               // total_count
    float kf    = (float)k;
    return lgammaf(kf + fs) - lgammaf(kf + 1.0f) - lgammaf(fs)
         + fs * log1pf(-ps) + kf * logf(ps);
}

// ---------------------------------------------------------------------------
// Kernel 1: build the 12 x 200 log-prob table in global workspace.
// ---------------------------------------------------------------------------
__global__ void build_table_kernel(const float* __restrict__ state_log_means,
                                   const float* __restrict__ state_log_frac_std,
                                   float* __restrict__ table) {
    for (int idx = threadIdx.x; idx < TABLE_N; idx += blockDim.x) {
        int s = idx / NK;
        int k = idx - s * NK;
        table[idx] = nb_logp(state_log_means[s], state_log_frac_std[s], k);
    }
}

// ---------------------------------------------------------------------------
// Shared hot loop: vectorized B128 gather over the aligned prefix + tiny tail.
// `table` is an (inlined) pointer to LDS.
// ---------------------------------------------------------------------------
__device__ __forceinline__ void emission_loop(const float* __restrict__ obs,
                                              const int*   __restrict__ states,
                                              const float* table,
                                              float* __restrict__ out,
                                              int total) {
    const int nchunks = total >> 2;                     // full float4 chunks
    const int stride  = (int)(blockDim.x * gridDim.x);
    const float4* obs4 = (const float4*)obs;
    const int4*   st4  = (const int4*)states;
    float4*       out4 = (float4*)out;

    for (int c = (int)(blockIdx.x * blockDim.x + threadIdx.x); c < nchunks; c += stride) {
        float4 o = obs4[c];
        int4   s = st4[c];
        float4 r;
        r.x = table[(s.x - 1) * NK + __float2int_rn(o.x)];
        r.y = table[(s.y - 1) * NK + __float2int_rn(o.y)];
        r.z = table[(s.z - 1) * NK + __float2int_rn(o.z)];
        r.w = table[(s.w - 1) * NK + __float2int_rn(o.w)];
        if (c == 0) r.x = 0.0f;                         // out[0] = 0
        out4[c] = r;
    }

    // tail: total % 4 elements (T = 2^24 + 2  ->  2 tail elems)
    if (blockIdx.x == 0 && threadIdx.x == 0) {
        for (int t = nchunks << 2; t < total; ++t) {
            float r = 0.0f;
            if (t > 0 && t < total - 1)
                r = table[(states[t] - 1) * NK + __float2int_rn(obs[t])];
            out[t] = r;                                  // out[T-1] = 0
        }
    }
}

// ---------------------------------------------------------------------------
// Kernel 2: stage table into LDS with CDNA5 async global->LDS B128 copies,
// then run the memory-bound gather.
// ---------------------------------------------------------------------------
__global__ void __launch_bounds__(256) nb_emission_kernel(
    const float* __restrict__ obs,
    const int*   __restrict__ states,
    const float* __restrict__ g_table,
    float* __restrict__ out,
    int total) {

    __shared__ __align__(16) float table[TABLE_N];

    // --- async-stage the 9.6KB table into LDS (GLOBAL_LOAD_ASYNC_TO_LDS_B128)
    {
        // Low 32 bits of a flat LDS pointer == LDS byte offset (ISA 10.2: upper
        // aperture bits are discarded on the LDS path).
        uint32_t lds_base = (uint32_t)(uintptr_t)(void*)table;
        const int nvec = TABLE_N / 4;                    // 600 x float4
        for (int i = threadIdx.x; i < nvec; i += (int)blockDim.x) {
            uint32_t ldsa = lds_base + (uint32_t)i * 16u;
            uint64_t ga   = (uint64_t)(uintptr_t)(g_table + i * 4);
            asm volatile("global_load_async_to_lds_b128 %0, %1, off"
                         :: "v"(ldsa), "v"(ga)
                         : "memory");
        }
        asm volatile("s_wait_asynccnt 0" ::: "memory");
    }
    __syncthreads();

    emission_loop(obs, states, table, out, total);
}

// ---------------------------------------------------------------------------
// Fallback (workspace too small): each block computes its own table in LDS.
// ---------------------------------------------------------------------------
__global__ void __launch_bounds__(256) nb_emission_fused_kernel(
    const float* __restrict__ obs,
    const int*   __restrict__ states,
    const float* __restrict__ state_log_means,
    const float* __restrict__ state_log_frac_std,
    float* __restrict__ out,
    int total) {

    __shared__ __align__(16) float table[TABLE_N];
    for (int idx = threadIdx.x; idx < TABLE_N; idx += (int)blockDim.x) {
        int s = idx / NK;
        int k = idx - s * NK;
        table[idx] = nb_logp(state_log_means[s], state_log_frac_std[s], k);
    }
    __syncthreads();

    emission_loop(obs, states, table, out, total);
}

// ---------------------------------------------------------------------------
// Host entry
// ---------------------------------------------------------------------------
extern "C" void kernel_launch(void* const* d_in, const int* in_sizes, int n_in,
                              void* d_out, int out_size, void* d_ws, size_t ws_size,
                              hipStream_t stream) {
    const float* obs    = (const float*)d_in[0];
    const int*   states = (const int*)d_in[1];
    const float* lm     = (const float*)d_in[2];
    const float* lf     = (const float*)d_in[3];
    float* out = (float*)d_out;

    int total = in_sizes[0];           // T = 16777218
    const int threads = 256;
    const int blocks  = 4096;          // grid-stride; ~4 float4 chunks/thread

    if (ws_size >= TABLE_N * sizeof(float) && d_ws != nullptr) {
        float* table = (float*)d_ws;
        build_table_kernel<<<1, 256, 0, stream>>>(lm, lf, table);
        nb_emission_kernel<<<blocks, threads, 0, stream>>>(obs, states, table, out, total);
    } else {
        nb_emission_fused_kernel<<<blocks, threads, 0, stream>>>(obs, states, lm, lf, out, total);
    }
}